// CTCLoss_52089363366256
// MI455X (gfx1250) — compile-verified
//
#include <hip/hip_runtime.h>
#include <stdint.h>

// Problem constants (match reference)
#define CTC_T 512
#define CTC_N 32
#define CTC_C 5000
#define CTC_S 100
#define CTC_L (2 * CTC_S + 1)   // 201 lattice positions
#define NEGF  (-1e30f)
#define DPF   8                 // async prefetch depth (emit ring buffer)

// ---------------------------------------------------------------------------
// CDNA5 async global->LDS gather (per-lane), tracked by ASYNCcnt.
// ---------------------------------------------------------------------------
#if defined(__has_builtin)
#if __has_builtin(__builtin_amdgcn_global_load_async_to_lds_b32)
#define CTC_ASYNC_BUILTIN 1
#endif
#if __has_builtin(__builtin_amdgcn_s_wait_asynccnt)
#define CTC_WAIT_BUILTIN 1
#endif
#endif

typedef __attribute__((address_space(1))) int ctc_gint;  // global i32
typedef __attribute__((address_space(3))) int ctc_lint;  // LDS i32

__device__ __forceinline__ void emit_gather_issue(const float* row, int lab,
                                                  float* lds_slot) {
#if defined(CTC_ASYNC_BUILTIN)
  __builtin_amdgcn_global_load_async_to_lds_b32(
      (ctc_gint*)(uintptr_t)(row + lab),
      (ctc_lint*)(uintptr_t)(lds_slot),
      /*offset=*/0, /*cpol=*/0);
#else
  // GVS mode: mem_addr = SGPR64 + VGPR_I32 + IOFFSET; VDST holds LDS byte addr.
  unsigned lds_off = (unsigned)(uintptr_t)lds_slot;   // low 32 bits = LDS offset
  unsigned goff    = (unsigned)(lab * 4);
  unsigned long long base = (unsigned long long)(uintptr_t)row;
  asm volatile("global_load_async_to_lds_b32 %0, %1, %2"
               :: "v"(lds_off), "v"(goff), "s"(base)
               : "memory");
#endif
}

template <int Ncnt>
__device__ __forceinline__ void wait_async_le() {
#if defined(CTC_WAIT_BUILTIN)
  __builtin_amdgcn_s_wait_asynccnt(Ncnt);
  asm volatile("" ::: "memory");
#else
  asm volatile("s_wait_asynccnt %0" :: "n"(Ncnt) : "memory");
#endif
}

__device__ __forceinline__ float lse3(float a, float b, float c) {
  float m = fmaxf(a, fmaxf(b, c));
  return m + __logf(__expf(a - m) + __expf(b - m) + __expf(c - m));
}

// ---------------------------------------------------------------------------
// One workgroup per batch row n. Thread l owns lattice position l.
// Alpha double-buffered in LDS; emits in a depth-8 async-filled LDS ring.
// ---------------------------------------------------------------------------
__global__ __launch_bounds__(256) void ctc_alpha_kernel(
    const float* __restrict__ lp,      // [T, N, C] log-probs
    const int*   __restrict__ tgt,     // [N, S]
    const int*   __restrict__ ilen_p,  // [N]
    const int*   __restrict__ tlen_p,  // [N]
    float*       __restrict__ per_n)   // [N] normalized per-row losses
{
  const int n   = blockIdx.x;
  const int tid = threadIdx.x;

  __shared__ float s_alpha[2][224];      // padded L
  __shared__ float s_emit[DPF][256];     // async gather landing zone (8 KB)

  const int tlen = tlen_p[n];
  const int ilen = ilen_p[n];            // uniform per block
  const int Lact = 2 * tlen + 1;

  // Per-thread lattice metadata (registers).
  int  lab      = 0;      // extended label: blank on even, target on odd
  bool can_skip = false;  // odd position, label != label two back
  if (tid < CTC_L) {
    if (tid & 1) {
      lab = tgt[n * CTC_S + (tid >> 1)];
      if (tid >= 3) {
        int prev = tgt[n * CTC_S + (tid >> 1) - 1];
        can_skip = (lab != prev);
      }
    }
  }
  const bool valid = (tid < CTC_L) && (tid < Lact);

  const float* lp_n = lp + (size_t)n * CTC_C;   // + t*N*C selects time step
  const size_t tstride = (size_t)CTC_N * CTC_C;

  // Prologue: issue emit gathers for rows 0..DPF-2.
  for (int k = 0; k < DPF - 1; ++k) {
    int r = (k < ilen) ? k : (ilen - 1);
    emit_gather_issue(lp_n + (size_t)r * tstride, lab, &s_emit[k][tid]);
  }

  // ---- step t = 0 (peeled: alpha init) --------------------------------
  {
    int r = (DPF - 1 < ilen) ? (DPF - 1) : (ilen - 1);
    emit_gather_issue(lp_n + (size_t)r * tstride, lab,
                      &s_emit[(DPF - 1) % DPF][tid]);
    wait_async_le<DPF - 1>();    // row 0 landed
    if (tid < CTC_L) {
      float e  = s_emit[0][tid];
      float nv = NEGF;
      if (tid == 0) nv = e;
      else if (tid == 1 && tlen > 0) nv = e;
      if (!valid) nv = NEGF;
      s_alpha[0][tid] = nv;
    }
    __syncthreads();
  }

  // ---- steps t = 1 .. ilen-1 (recurrence) -----------------------------
  for (int t = 1; t < ilen; ++t) {
    {
      int pr = t + DPF - 1;
      int r  = (pr < ilen) ? pr : (ilen - 1);
      emit_gather_issue(lp_n + (size_t)r * tstride, lab,
                        &s_emit[pr % DPF][tid]);
    }
    // Oldest (row t) is guaranteed complete once <= DPF-1 remain outstanding.
    wait_async_le<DPF - 1>();

    if (tid < CTC_L) {
      float e  = s_emit[t % DPF][tid];
      const float* ap = s_alpha[(t - 1) & 1];
      float a0 = ap[tid];
      float a1 = (tid >= 1) ? ap[tid - 1] : NEGF;
      float a2 = can_skip ? ap[tid - 2] : NEGF;   // can_skip implies tid>=3
      float nv = lse3(a0, a1, a2) + e;
      if (!valid) nv = NEGF;
      s_alpha[t & 1][tid] = nv;
    }
    __syncthreads();   // publish alpha[t] before step t+1 reads it
  }

  if (tid == 0) {
    const float* af = s_alpha[(ilen - 1) & 1];
    int   last = 2 * tlen;
    int   li   = (last - 1 > 0) ? (last - 1) : 0;
    float ab   = af[last];
    float al   = af[li];
    float m    = fmaxf(ab, al);
    float ll   = m + __logf(__expf(ab - m) + __expf(al - m));
    float loss = -ll;
    // zero_infinity semantics: isfinite(loss) && loss < 1e29 else 0
    if (!__builtin_isfinite(loss) || !(loss < 1e29f)) loss = 0.0f;
    float denom = (float)((tlen > 0) ? tlen : 1);
    per_n[n] = loss / denom;
  }
}

// Fixed-order serial reduction -> deterministic scalar mean.
__global__ void ctc_reduce_kernel(const float* __restrict__ per_n,
                                  float* __restrict__ out) {
  if (threadIdx.x == 0 && blockIdx.x == 0) {
    float s = 0.0f;
    for (int i = 0; i < CTC_N; ++i) s += per_n[i];
    out[0] = s * (1.0f / (float)CTC_N);
  }
}

extern "C" void kernel_launch(void* const* d_in, const int* in_sizes, int n_in,
                              void* d_out, int out_size, void* d_ws, size_t ws_size,
                              hipStream_t stream) {
  const float* lp   = (const float*)d_in[0];   // [T,N,C] f32 log-probs
  const int*   tgt  = (const int*)d_in[1];     // [N,S]
  const int*   ilen = (const int*)d_in[2];     // [N]
  const int*   tlen = (const int*)d_in[3];     // [N]
  float* per_n = (float*)d_ws;                 // 32 floats of scratch

  ctc_alpha_kernel<<<CTC_N, 256, 0, stream>>>(lp, tgt, ilen, tlen, per_n);
  ctc_reduce_kernel<<<1, 64, 0, stream>>>(per_n, (float*)d_out);
}